// Retinanet_38628935860248
// MI455X (gfx1250) — compile-verified
//
#include <hip/hip_runtime.h>
#include <hip/hip_bf16.h>
#include <math.h>

// ---------------- problem constants (from reference) ----------------
constexpr int   Bsz       = 8;
constexpr int   Csz       = 80;
constexpr int   Ksz       = 10;
constexpr int   MAXN      = 76725;          // 9 * (80^2+40^2+20^2+10^2+5^2)
constexpr float SCORE_THR = 0.005f;
constexpr float IOU_THR   = 0.05f;
constexpr float STD_      = 0.2f;
constexpr float NEGV      = -1000000000.0f; // exactly representable in fp32
constexpr float IMGSZ     = 640.0f;
constexpr float MIN_SIZE  = 0.01f;
constexpr float BBOX_CLIP = 4.135166556742356f; // log(1000/16)

#define BDIM_NMS 1024  // 32 wave32s per workgroup

// ---------------- CDNA5 async global->LDS gather helpers ----------------
#define AS1 __attribute__((address_space(1)))
#define AS3 __attribute__((address_space(3)))

#if defined(__has_builtin)
#if __has_builtin(__builtin_amdgcn_global_load_async_to_lds_b32)
#define HAVE_ASYNC_LDS 1
#endif
#if __has_builtin(__builtin_amdgcn_s_wait_asynccnt)
#define HAVE_WAIT_ASYNC 1
#endif
#endif

__device__ __forceinline__ void async_gather_b32(const float* g, float* l) {
#ifdef HAVE_ASYNC_LDS
    // builtin signature (per hipcc diagnostic): (global int*, shared int*, imm offset, cpol)
    __builtin_amdgcn_global_load_async_to_lds_b32(
        (AS1 int*)(g), (AS3 int*)(l), 0, 0);
#else
    *l = *g;  // fallback: plain load + LDS store
#endif
}

__device__ __forceinline__ void wait_async_lds() {
#ifdef HAVE_ASYNC_LDS
#ifdef HAVE_WAIT_ASYNC
    __builtin_amdgcn_s_wait_asynccnt(0);
#else
    asm volatile("s_wait_asynccnt 0" ::: "memory");
#endif
#endif
}

// ---------------- kernel 1: decode + clip + size flag ----------------
__global__ void retinanet_decode(const float* __restrict__ deltas,   // (B,N,4)
                                 const float* __restrict__ anchors,  // (N,4)
                                 float*       __restrict__ boxes,    // ws: (B,N,4)
                                 unsigned char* __restrict__ bigf,   // ws: (B,N)
                                 int N) {
    int n = blockIdx.x * blockDim.x + threadIdx.x;
    int b = blockIdx.y;
    if (n >= N) return;

    float4 a = ((const float4*)anchors)[n];
    float4 d = ((const float4*)deltas)[(size_t)b * N + n];

    float wa  = a.z - a.x;
    float ha  = a.w - a.y;
    float cxa = a.x + 0.5f * wa;
    float cya = a.y + 0.5f * ha;

    float dx = d.x * STD_, dy = d.y * STD_;
    float dw = d.z * STD_, dh = d.w * STD_;

    float cx = dx * wa + cxa;
    float cy = dy * ha + cya;
    float w  = expf(fminf(dw, BBOX_CLIP)) * wa;   // upper clip only (a_max)
    float h  = expf(fminf(dh, BBOX_CLIP)) * ha;

    float x1 = fminf(fmaxf(cx - 0.5f * w, 0.0f), IMGSZ);
    float y1 = fminf(fmaxf(cy - 0.5f * h, 0.0f), IMGSZ);
    float x2 = fminf(fmaxf(cx + 0.5f * w, 0.0f), IMGSZ);
    float y2 = fminf(fmaxf(cy + 0.5f * h, 0.0f), IMGSZ);

    ((float4*)boxes)[(size_t)b * N + n] = make_float4(x1, y1, x2, y2);
    bigf[(size_t)b * N + n] =
        ((x2 - x1 >= MIN_SIZE) && (y2 - y1 >= MIN_SIZE)) ? 1 : 0;
}

// ---------------- kernel 2: per-(b,c) NMS with LDS-resident scores ----------------
// One workgroup per (b,c). Scores gathered (async) into 320KB CDNA5 LDS once;
// all K argmax rounds scan LDS. Boxes re-read from L2-resident workspace.
__global__ __launch_bounds__(BDIM_NMS, 1)
void retinanet_nms(const float* __restrict__ cls,     // (B,N,C)
                   const float* __restrict__ boxes,   // (B,N,4)
                   const unsigned char* __restrict__ bigf, // (B,N)
                   float* __restrict__ out, int N) {
    __shared__ float e[MAXN];          // ~300 KB: masked scores, mutated in place
    __shared__ float rv[32];
    __shared__ int   ri[32];
    __shared__ float selbox[4];
    __shared__ int   seln_sh;
    __shared__ int   sel_ks[Ksz];

    const int tid = threadIdx.x;
    const int c = blockIdx.x, b = blockIdx.y;
    const float* cls_bc = cls + (size_t)b * N * Csz + c;
    const unsigned char* big_b = bigf + (size_t)b * N;
    const float4* box_b = (const float4*)boxes + (size_t)b * N;

    // Stage strided score gather directly into LDS (async on CDNA5).
    for (int n = tid; n < N; n += BDIM_NMS)
        async_gather_b32(cls_bc + (size_t)n * Csz, &e[n]);
    wait_async_lds();
    __syncthreads();

    // Apply score-threshold + min-size mask in place (e becomes "masked").
    for (int n = tid; n < N; n += BDIM_NMS) {
        float s = e[n];
        e[n] = (s > SCORE_THR && big_b[n]) ? s : NEGV;
    }
    __syncthreads();

    for (int k = 0; k < Ksz; ++k) {
        // ---- argmax over e with first-index tie-break ----
        float bv = -INFINITY;
        int   bn = 0x7fffffff;
        for (int n = tid; n < N; n += BDIM_NMS) {
            float v = e[n];
            if (v > bv || (v == bv && n < bn)) { bv = v; bn = n; }
        }
        #pragma unroll
        for (int off = 16; off > 0; off >>= 1) {   // wave32 butterfly
            float ov = __shfl_xor(bv, off, 32);
            int   on = __shfl_xor(bn, off, 32);
            if (ov > bv || (ov == bv && on < bn)) { bv = ov; bn = on; }
        }
        if ((tid & 31) == 0) { rv[tid >> 5] = bv; ri[tid >> 5] = bn; }
        __syncthreads();
        if (tid < 32) {
            bv = rv[tid]; bn = ri[tid];
            #pragma unroll
            for (int off = 16; off > 0; off >>= 1) {
                float ov = __shfl_xor(bv, off, 32);
                int   on = __shfl_xor(bn, off, 32);
                if (ov > bv || (ov == bv && on < bn)) { bv = ov; bn = on; }
            }
            if (tid == 0) {
                seln_sh   = bn;
                sel_ks[k] = bn;
                float4 sb = box_b[bn];
                selbox[0] = sb.x; selbox[1] = sb.y;
                selbox[2] = sb.z; selbox[3] = sb.w;
            }
        }
        __syncthreads();

        // ---- suppression: e[n] = NEG where IoU(sel, n) > thr; e[sel] = NEG ----
        const int   sn  = seln_sh;
        const float sx1 = selbox[0], sy1 = selbox[1];
        const float sx2 = selbox[2], sy2 = selbox[3];
        const float sA  = (sx2 - sx1) * (sy2 - sy1);
        for (int n = tid; n < N; n += BDIM_NMS) {
            float v = e[n];
            if (v == NEGV) continue;          // already dead: idempotent skip
            if (n == sn) { e[n] = NEGV; continue; }
            float4 bx = box_b[n];
            float ix1 = fmaxf(sx1, bx.x), iy1 = fmaxf(sy1, bx.y);
            float ix2 = fminf(sx2, bx.z), iy2 = fminf(sy2, bx.w);
            float inter = fmaxf(ix2 - ix1, 0.0f) * fmaxf(iy2 - iy1, 0.0f);
            float area  = (bx.z - bx.x) * (bx.w - bx.y);
            float iou   = inter / (sA + area - inter + 1e-9f);
            if (iou > IOU_THR) e[n] = NEGV;
        }
        __syncthreads();
    }

    // ---- outputs: validity from ORIGINAL mask at selected index ----
    if (tid < Ksz) {
        const int k = tid;
        const int n = sel_ks[k];
        float s = cls_bc[(size_t)n * Csz];
        bool valid = (s > SCORE_THR) && big_b[n];
        float4 bx = box_b[n];
        if (!valid) { bx = make_float4(0.f, 0.f, 0.f, 0.f); s = 0.f; }
        const size_t bc = (size_t)b * Csz + c;
        float* ob = out + (bc * Ksz + k) * 4;
        ob[0] = bx.x; ob[1] = bx.y; ob[2] = bx.z; ob[3] = bx.w;
        out[(size_t)Bsz * Csz * Ksz * 4 + bc * Ksz + k] = s;                 // scores
        out[(size_t)Bsz * Csz * Ksz * 5 + bc * Ksz + k] = valid ? 1.f : 0.f; // valid
    }
}

// ---------------- launcher ----------------
extern "C" void kernel_launch(void* const* d_in, const int* in_sizes, int n_in,
                              void* d_out, int out_size, void* d_ws, size_t ws_size,
                              hipStream_t stream) {
    const float* pred_cls = (const float*)d_in[0];   // (B,N,C)
    const float* deltas   = (const float*)d_in[1];   // (B,N,4)
    const float* anchors  = (const float*)d_in[2];   // (N,4)
    int N = (n_in >= 3) ? (in_sizes[2] / 4) : MAXN;
    if (N > MAXN) N = MAXN;

    // workspace: boxes (B*N*16 B) then big flags (B*N bytes)
    float* boxes = (float*)d_ws;
    unsigned char* bigf = (unsigned char*)d_ws + (size_t)Bsz * N * 16;

    dim3 g1((N + 255) / 256, Bsz);
    retinanet_decode<<<g1, 256, 0, stream>>>(deltas, anchors, boxes, bigf, N);

    dim3 g2(Csz, Bsz);   // one workgroup per (class, batch)
    retinanet_nms<<<g2, BDIM_NMS, 0, stream>>>(pred_cls, boxes, bigf,
                                               (float*)d_out, N);
}